// LGCN_Encoder_53326313947473
// MI455X (gfx1250) — compile-verified
//
#include <hip/hip_runtime.h>
#include <hip/hip_bf16.h>

// LightGCN encoder for MI455X (gfx1250, wave32).
//
// ego0 = concat(user_emb, item_emb)            [N,64], N = 250k
// repeat 3x:  ego = SpMM(A, ego); acc += ego   (A: COO, 5M nnz)
// mean = acc/4 ; outputs = (mean split, ego0 split, prototypes)
//
// Roofline: ~128 FLOPs per 524 B of random traffic per nnz -> purely
// memory/atomic bound. The 64MB embedding matrix fits in the 192MB L2, so
// the strategy is: keep x/y/acc L2-resident (regular temporal loads/stores,
// f32 atomics serviced by L2) and mark the streamed-once edge lists and the
// write-once ego0 output non-temporal so they don't evict the hot set.

#define USER_NUM 100000
#define ITEM_NUM 150000
#define NTOT     (USER_NUM + ITEM_NUM)   // 250000
#define EMB      64
#define NNZ_CNT  5000000
#define PROTO    2048

// native clang vector type: required by __builtin_nontemporal_* builtins
typedef float v4f __attribute__((ext_vector_type(4)));

// ---- init: acc = ego0, ego0_out = ego0 (NT store), x = ego0 ----
__global__ void lgcn_init_kernel(const v4f* __restrict__ ue,
                                 const v4f* __restrict__ ie,
                                 v4f* __restrict__ acc,
                                 v4f* __restrict__ ego0_out,
                                 v4f* __restrict__ x) {
    const size_t uN = (size_t)USER_NUM * EMB / 4;
    const size_t tN = (size_t)NTOT * EMB / 4;
    size_t i = (size_t)blockIdx.x * blockDim.x + threadIdx.x;
    if (i >= tN) return;
    v4f v = (i < uN) ? ue[i] : ie[i - uN];
    acc[i] = v;                              // re-read by accum: keep temporal
    x[i]   = v;                              // re-read by spmm:  keep temporal
    __builtin_nontemporal_store(v, &ego0_out[i]);  // written once, never re-read
}

// ---- SpMM: one wave32 per nonzero; lane handles 2 of the 64 floats ----
// y[row][:] += val * x[col][:]   via non-returning global_atomic_add_f32
__global__ void lgcn_spmm_kernel(const float* __restrict__ vals,
                                 const int*   __restrict__ rows,
                                 const int*   __restrict__ cols,
                                 const float* __restrict__ x,
                                 float*       __restrict__ y) {
    unsigned wid  = blockIdx.x * (blockDim.x >> 5) + (threadIdx.x >> 5);
    unsigned lane = threadIdx.x & 31u;
    if (wid >= NNZ_CNT) return;

    // Edge metadata: streamed exactly once per layer -> non-temporal loads
    // (TH=NT) so the 60MB of index/value arrays don't evict the L2-resident
    // embedding tables. wid is wave-uniform -> scalarize row/col so the
    // gather and the atomics use SGPR bases (saddr-form global ops).
    int   r = __builtin_amdgcn_readfirstlane(__builtin_nontemporal_load(&rows[wid]));
    int   c = __builtin_amdgcn_readfirstlane(__builtin_nontemporal_load(&cols[wid]));
    float v = __builtin_nontemporal_load(&vals[wid]);

    // coalesced 256B gather of x[c][0:64]: one global_load_b64 per lane
    float2 d = ((const float2*)(x + (size_t)c * EMB))[lane];

    float* yp = y + (size_t)r * EMB + (size_t)lane * 2;
    // fire-and-forget f32 atomics (L2-resident target rows, STOREcnt-tracked)
    unsafeAtomicAdd(yp,     v * d.x);
    unsafeAtomicAdd(yp + 1, v * d.y);
}

// ---- acc = (acc + y) * s   (s = 1.0 for inner layers, 0.25 on the last:
//      fuses the mean division into the final accumulate pass) ----
__global__ void lgcn_accum_kernel(v4f* __restrict__ acc,
                                  const v4f* __restrict__ y,
                                  float s) {
    const size_t tN = (size_t)NTOT * EMB / 4;
    size_t i = (size_t)blockIdx.x * blockDim.x + threadIdx.x;
    if (i >= tN) return;
    v4f a = acc[i];
    v4f b = y[i];
    a = (a + b) * s;
    acc[i] = a;
}

extern "C" void kernel_launch(void* const* d_in, const int* in_sizes, int n_in,
                              void* d_out, int out_size, void* d_ws, size_t ws_size,
                              hipStream_t stream) {
    const float* user_emb = (const float*)d_in[0];  // [100000,64]
    const float* item_emb = (const float*)d_in[1];  // [150000,64]
    const float* uproto   = (const float*)d_in[2];  // [2048,64]
    const float* iproto   = (const float*)d_in[3];  // [2048,64]
    const float* vals     = (const float*)d_in[4];  // [5M]
    const int*   rows     = (const int*)  d_in[5];  // [5M]
    const int*   cols     = (const int*)  d_in[6];  // [5M]

    float* out = (float*)d_out;
    const size_t tE = (size_t)NTOT * EMB;           // 16,000,000 floats

    // d_out layout (flat, return order):
    // [0, tE)       mean_emb  (user_all ++ item_all, contiguous [N,64])
    // [tE, 2tE)     ego0      (user_cl ++ item_cl)
    // [2tE, ...)    user_prototypes, item_prototypes
    float* acc      = out;            // accumulator lives in the output
    float* ego0_out = out + tE;
    float* uproto_o = out + 2 * tE;
    float* iproto_o = out + 2 * tE + (size_t)PROTO * EMB;

    // workspace: ping/pong embedding buffers (2 x 64MB)
    float* x = (float*)d_ws;
    float* y = x + tE;

    const int TPB = 256;

    // prototypes pass straight through
    (void)hipMemcpyAsync(uproto_o, uproto, (size_t)PROTO * EMB * sizeof(float),
                         hipMemcpyDeviceToDevice, stream);
    (void)hipMemcpyAsync(iproto_o, iproto, (size_t)PROTO * EMB * sizeof(float),
                         hipMemcpyDeviceToDevice, stream);

    // init acc/ego0/x from concat(user_emb, item_emb)
    const unsigned elem_blocks = (unsigned)((tE / 4 + TPB - 1) / TPB);
    lgcn_init_kernel<<<elem_blocks, TPB, 0, stream>>>(
        (const v4f*)user_emb, (const v4f*)item_emb,
        (v4f*)acc, (v4f*)ego0_out, (v4f*)x);

    // 3 propagation layers
    const unsigned waves_per_block = TPB / 32;                      // 8
    const unsigned spmm_blocks = (NNZ_CNT + waves_per_block - 1) / waves_per_block;

    for (int layer = 0; layer < 3; ++layer) {
        (void)hipMemsetAsync(y, 0, tE * sizeof(float), stream);
        lgcn_spmm_kernel<<<spmm_blocks, TPB, 0, stream>>>(vals, rows, cols, x, y);
        // last layer folds the /4 mean into the accumulate pass
        const float s = (layer == 2) ? 0.25f : 1.0f;
        lgcn_accum_kernel<<<elem_blocks, TPB, 0, stream>>>(
            (v4f*)acc, (const v4f*)y, s);
        // next layer reads this layer's result
        float* t = x; x = y; y = t;
    }
}